// NSGT_sliced_63084479643654
// MI455X (gfx1250) — compile-verified
//
#include <hip/hip_runtime.h>
#include <hip/hip_bf16.h>
#include <stdint.h>

// NSGT arrange(fwd): per-band circular row shift. Pure permuted memcpy,
// 1.41 GB traffic @ 23.3 TB/s -> ~60us floor. Strategy:
//  - bands 1..3: direct b128 NT copy, 8 loads in flight per thread
//  - band 0:     gfx1250 async global->LDS->global DMA path (ASYNCcnt)

typedef float v4f __attribute__((ext_vector_type(4)));

#define LDS_AS __attribute__((address_space(3)))

static __device__ __forceinline__ unsigned lds_off_u32(void* p) {
    // generic -> local addrspacecast, then ptrtoint: yields the 32-bit LDS
    // byte offset (flat LDS aperture keeps the offset in the low 32 bits).
    return (unsigned)(unsigned long long)(LDS_AS void*)p;
}

// ---------------------------------------------------------------------------
// Direct path: one float4 per lane-iteration, 8 iterations batched so each
// wave keeps 8 x 512B loads outstanding before storing (MLP for 23.3 TB/s).
// Grid is sized so blocks tile the band exactly: no bounds checks, full EXEC.
//   idx layout (float4 units): idx = ((slice*C*B + cb) << LOG_MV) | jv
//   parity(slice) picks midv; src row offset = (jv + midv) & (Mv-1).
// ---------------------------------------------------------------------------
template <int LOG_MV, int LOG_CB>
__global__ __launch_bounds__(256) void roll_band_direct(const v4f* __restrict__ in,
                                                        v4f* __restrict__ out) {
    constexpr unsigned MV   = 1u << LOG_MV;
    constexpr unsigned MASK = MV - 1u;
    constexpr unsigned Q    = MV >> 2;  // Mv/4

    const unsigned base = blockIdx.x * 2048u + threadIdx.x;  // 256 thr * 8 unroll

    v4f v[8];
#pragma unroll
    for (int u = 0; u < 8; ++u) {
        const unsigned idx    = base + (unsigned)u * 256u;
        const unsigned parity = (idx >> (LOG_MV + LOG_CB)) & 1u;  // slice & 1
        const unsigned midv   = parity ? Q : 3u * Q;              // M/4 vs 3M/4
        const unsigned src    = (idx & ~MASK) | ((idx + midv) & MASK);
        v[u] = __builtin_nontemporal_load(in + src);  // th:TH_LOAD_NT, b128
    }
#pragma unroll
    for (int u = 0; u < 8; ++u) {
        const unsigned idx = base + (unsigned)u * 256u;
        __builtin_nontemporal_store(v[u], out + idx);  // th:TH_STORE_NT, b128
    }
}

// ---------------------------------------------------------------------------
// Band 0 via the CDNA5 async DMA path. Shape (128,2,32,1024): 8192 rows of
// 4KB. One row per wave: 8 x global_load_async_to_lds_b128 (rotation folded
// into the *source* address so LDS receives the already-rolled row),
// s_wait_asynccnt 0, 8 x global_store_async_from_lds_b128, s_wait_asynccnt 0.
// Shift amounts (256 / 768 floats) are multiples of 4, so each 16B block is
// contiguous in the source (never straddles the wrap).
// ---------------------------------------------------------------------------
__global__ __launch_bounds__(256) void roll_band0_async(const float* __restrict__ in,
                                                        float* __restrict__ out,
                                                        unsigned nrows) {
    __shared__ __align__(16) float lds[8][1024];  // 4KB per wave, 32KB/block

    const unsigned lane    = threadIdx.x & 31u;
    const unsigned wave    = threadIdx.x >> 5;
    const unsigned waveId  = blockIdx.x * 8u + wave;
    const unsigned nWaves  = gridDim.x * 8u;
    const unsigned ldsBase = lds_off_u32(&lds[wave][0]);

    for (unsigned row = waveId; row < nrows; row += nWaves) {
        const unsigned slice = row >> 6;  // C*B = 64 rows per slice
        const unsigned mid   = (slice & 1u) ? 256u : 768u;  // M/4 : 3M/4 (floats)
        const float* rowin   = in + (size_t)row * 1024u;
        float* rowout        = out + (size_t)row * 1024u;

#pragma unroll
        for (int c = 0; c < 8; ++c) {
            const unsigned o      = (unsigned)c * 128u + lane * 4u;  // out offset (floats)
            const unsigned so     = (o + mid) & 1023u;               // rolled source
            const unsigned ldsoff = ldsBase + o * 4u;
            const float* gsrc     = rowin + so;
            asm volatile("global_load_async_to_lds_b128 %0, %1, off"
                         :
                         : "v"(ldsoff), "v"(gsrc)
                         : "memory");
        }
        __builtin_amdgcn_s_wait_asynccnt(0);  // loads landed in LDS

#pragma unroll
        for (int c = 0; c < 8; ++c) {
            const unsigned o      = (unsigned)c * 128u + lane * 4u;
            const unsigned ldsoff = ldsBase + o * 4u;
            float* gdst           = rowout + o;
            asm volatile("global_store_async_from_lds_b128 %0, %1, off"
                         :
                         : "v"(gdst), "v"(ldsoff)
                         : "memory");
        }
        __builtin_amdgcn_s_wait_asynccnt(0);  // LDS buffer reusable
    }
}

// ---------------------------------------------------------------------------
// Launch. Output = 4 bands concatenated flat in return order.
//   band0:  8,388,608 f32   band1: 33,554,432   band2: 67,108,864   band3: 67,108,864
// float4 counts: 2,097,152 / 8,388,608 / 16,777,216 / 16,777,216 — all exact
// multiples of 2048 (256 threads * 8 unroll), so grids tile exactly.
// ---------------------------------------------------------------------------
extern "C" void kernel_launch(void* const* d_in, const int* in_sizes, int n_in,
                              void* d_out, int out_size, void* d_ws, size_t ws_size,
                              hipStream_t stream) {
    (void)in_sizes; (void)n_in; (void)out_size; (void)d_ws; (void)ws_size;

    const float* in0 = (const float*)d_in[0];
    const float* in1 = (const float*)d_in[1];
    const float* in2 = (const float*)d_in[2];
    const float* in3 = (const float*)d_in[3];
    float* out = (float*)d_out;

    float* out0 = out;
    float* out1 = out + 8388608u;    // + band0
    float* out2 = out + 41943040u;   // + band0+band1
    float* out3 = out + 109051904u;  // + band0+band1+band2

    // band0: (128,2,32,1024) -> 8192 rows, one row per wave (1024 blocks * 8 waves)
    roll_band0_async<<<dim3(1024), dim3(256), 0, stream>>>(in0, out0, 8192u);

    // band1: (128,2,64,2048): Mv=512 (LOG_MV=9), C*B=128 (LOG_CB=7), 8,388,608/2048 blocks
    roll_band_direct<9, 7><<<dim3(4096), dim3(256), 0, stream>>>((const v4f*)in1, (v4f*)out1);

    // band2: (128,2,64,4096): Mv=1024 (10), C*B=128 (7), 16,777,216/2048 blocks
    roll_band_direct<10, 7><<<dim3(8192), dim3(256), 0, stream>>>((const v4f*)in2, (v4f*)out2);

    // band3: (128,2,32,8192): Mv=2048 (11), C*B=64 (6)
    roll_band_direct<11, 6><<<dim3(8192), dim3(256), 0, stream>>>((const v4f*)in3, (v4f*)out3);
}